// ReadHead_52407190946070
// MI455X (gfx1250) — compile-verified
//
#include <hip/hip_runtime.h>
#include <math.h>

#define EPSV  1e-8f
#define NH    8
#define WRD   128
#define NC    16384
#define NB    16
#define LDSS  132          // padded LDS row stride (floats): conflict-free fragment reads

typedef __attribute__((ext_vector_type(2))) float v2f;
typedef __attribute__((ext_vector_type(8))) float v8f;
typedef __attribute__((ext_vector_type(4))) int   v4i;

// ---------------- CDNA5 async global->LDS copy (ASYNCcnt-tracked) ----------------
#if __has_builtin(__builtin_amdgcn_global_load_async_to_lds_b128)
#define HAVE_ASYNC_LDS 1
#endif

__device__ __forceinline__ void copy16_async(const float* g, float* l) {
#ifdef HAVE_ASYNC_LDS
  __builtin_amdgcn_global_load_async_to_lds_b128(
      (__attribute__((address_space(1))) v4i*)g,
      (__attribute__((address_space(3))) v4i*)l, 0, 0);
#else
  *(float4*)l = *(const float4*)g;
#endif
}

__device__ __forceinline__ void async_join() {
#ifdef HAVE_ASYNC_LDS
#if __has_builtin(__builtin_amdgcn_s_wait_asynccnt)
  __builtin_amdgcn_s_wait_asynccnt(0);
#else
  asm volatile("s_wait_asynccnt 0x0" ::: "memory");
#endif
#endif
  __syncthreads();
}

// ---------------- block reduction helpers (wave32) ----------------
__device__ __forceinline__ float waveRed(float v, bool isSum) {
#pragma unroll
  for (int o = 16; o; o >>= 1) {
    float other = __shfl_xor(v, o, 32);
    v = isSum ? (v + other) : fmaxf(v, other);
  }
  return v;
}

__device__ float blockRed(float v, float* red, bool isSum) {
  v = waveRed(v, isSum);
  int lane = threadIdx.x & 31, w = threadIdx.x >> 5;
  int nw = (blockDim.x + 31) >> 5;
  if (lane == 0) red[w] = v;
  __syncthreads();
  if (w == 0) {
    float x = (lane < nw) ? red[lane] : (isSum ? 0.0f : -INFINITY);
    x = waveRed(x, isSum);
    if (lane == 0) red[32] = x;
  }
  __syncthreads();
  float r = red[32];
  __syncthreads();
  return r;
}

__device__ __forceinline__ float softplusf(float x) {
  return (x > 20.0f) ? x : log1pf(__expf(x));
}

// ---------------- kernel 1: control activations ----------------
__global__ void ntm_prep_kernel(const float* __restrict__ controls,
                                float* __restrict__ keysT,
                                float* __restrict__ params) {
  int t = threadIdx.x;
  int b = t >> 4, hp = t & 15;
  float* krow = keysT + (size_t)(b * 16 + hp) * WRD;
  if (hp >= NH) {           // zero-pad fake heads for the 16-wide WMMA N dim
    for (int w = 0; w < WRD; ++w) krow[w] = 0.0f;
    return;
  }
  const float* base = controls + (size_t)b * (NH * (WRD + 6));
  float ss = 0.0f;
  for (int w = 0; w < WRD; ++w) {
    float v = tanhf(base[hp * WRD + w]);
    krow[w] = v;
    ss += v * v;
  }
  const float* p = base + NH * WRD;
  float beta  = softplusf(p[hp]);
  float gate  = 1.0f / (1.0f + __expf(-p[NH + hp]));
  float s0 = p[2 * NH + hp * 3 + 0];
  float s1 = p[2 * NH + hp * 3 + 1];
  float s2 = p[2 * NH + hp * 3 + 2];
  float m  = fmaxf(s0, fmaxf(s1, s2));
  float e0 = __expf(s0 - m), e1 = __expf(s1 - m), e2 = __expf(s2 - m);
  float es = e0 + e1 + e2;
  float gamma = 1.0f + softplusf(p[5 * NH + hp]);
  float* pr = params + (size_t)(b * NH + hp) * 8;
  pr[0] = beta; pr[1] = gate;
  pr[2] = e0 / es; pr[3] = e1 / es; pr[4] = e2 / es;
  pr[5] = gamma; pr[6] = sqrtf(ss);
}

// ---------------- kernel 2: prev_weights = softmax(bias) per head ----------------
__global__ void ntm_prevw_kernel(const float* __restrict__ bias,
                                 float* __restrict__ prevW) {
  __shared__ float red[33];
  int h = blockIdx.x;
  const float* row = bias + (size_t)h * NC;
  float mx = -INFINITY;
  for (int c = threadIdx.x; c < NC; c += blockDim.x) mx = fmaxf(mx, row[c]);
  mx = blockRed(mx, red, false);
  float s = 0.0f;
  for (int c = threadIdx.x; c < NC; c += blockDim.x) s += __expf(row[c] - mx);
  s = blockRed(s, red, true);
  float inv = 1.0f / s;
  for (int c = threadIdx.x; c < NC; c += blockDim.x)
    prevW[(size_t)h * NC + c] = __expf(row[c] - mx) * inv;
}

// ---------------- kernel 3: dots = memory x keysT via WMMA from LDS ------------
// block = 128 threads (4 waves); stages 64 memory rows + keys into LDS with
// async b128 copies, then each wave computes a 16(c) x 16(h) f32 WMMA tile.
__global__ void ntm_dots_kernel(const float* __restrict__ mem,
                                const float* __restrict__ keysT,
                                float* __restrict__ dots,
                                float* __restrict__ mnorm) {
  __shared__ __align__(16) float smem[64 * LDSS];   // 33.8 KB
  __shared__ __align__(16) float skey[16 * LDSS];   //  8.4 KB
  int tid = threadIdx.x;
  int b  = blockIdx.x >> 8;            // 256 row-groups of 64 per batch
  int c0base = (blockIdx.x & 255) << 6;

  // stage 64 x 128 f32 memory tile: 2048 x 16B chunks, coalesced
  const float* gbase = mem + ((size_t)b * NC + c0base) * WRD;
#pragma unroll
  for (int i = 0; i < 16; ++i) {
    int ci = tid + i * 128;
    int row = ci >> 5, ch = ci & 31;
    copy16_async(gbase + row * WRD + ch * 4, smem + row * LDSS + ch * 4);
  }
  // stage 16 x 128 keys tile: 512 x 16B chunks
  const float* kbase = keysT + (size_t)b * 16 * WRD;
#pragma unroll
  for (int i = 0; i < 4; ++i) {
    int ci = tid + i * 128;
    int row = ci >> 5, ch = ci & 31;
    copy16_async(kbase + row * WRD + ch * 4, skey + row * LDSS + ch * 4);
  }
  async_join();

  int wv = tid >> 5, lane = tid & 31;
  int lm = lane & 15, hi = lane >> 4;
  int c0 = c0base + wv * 16;
  const float* arow = smem + (wv * 16 + lm) * LDSS + 2 * hi;  // A: M = c-row
  const float* brow = skey + lm * LDSS + 2 * hi;              // B: N = head
  v8f acc = {0.f, 0.f, 0.f, 0.f, 0.f, 0.f, 0.f, 0.f};
  float ss = 0.0f;
#pragma unroll 4
  for (int k0 = 0; k0 < WRD; k0 += 4) {
    v2f a  = *(const v2f*)(arow + k0);
    v2f bb = *(const v2f*)(brow + k0);
    ss += a.x * a.x + a.y * a.y;
    acc = __builtin_amdgcn_wmma_f32_16x16x4_f32(false, a, false, bb,
                                                (short)0, acc, false, false);
  }
  // combine the two half-wave partial sums-of-squares of each memory row
  ss += __shfl_xor(ss, 16, 32);
  if (lane < 16) mnorm[(size_t)b * NC + c0 + lane] = sqrtf(ss);

  // D layout: lane holds N=lm (head), rows M = v + 8*hi -> dots[b][c][h16]
  float* dbase = dots + ((size_t)b * NC + c0 + 8 * hi) * 16 + lm;
#pragma unroll
  for (int v = 0; v < 8; ++v) dbase[v * 16] = acc[v];
}

// ---------------- kernel 4: softmax / gate / shift / sharpen -> read_dist ------
__global__ void ntm_weights_kernel(const float* __restrict__ dots,
                                   const float* __restrict__ mnorm,
                                   const float* __restrict__ params,
                                   const float* __restrict__ prevW,
                                   float* __restrict__ dist) {
  __shared__ float red[33];
  int b = blockIdx.x >> 3, h = blockIdx.x & 7;
  const float* pr = params + (size_t)(b * NH + h) * 8;
  float beta = pr[0], gate = pr[1], s0 = pr[2], s1 = pr[3], s2 = pr[4];
  float gamma = pr[5], knorm = pr[6];
  const float* dcol = dots + (size_t)b * NC * 16 + h;
  const float* mn   = mnorm + (size_t)b * NC;
  const float* pw   = prevW + (size_t)h * NC;

  float mx = -INFINITY;
  for (int c = threadIdx.x; c < NC; c += blockDim.x) {
    float sc = dcol[(size_t)c * 16] / (knorm * mn[c] + EPSV);
    mx = fmaxf(mx, sc * beta);
  }
  mx = blockRed(mx, red, false);
  float s = 0.0f;
  for (int c = threadIdx.x; c < NC; c += blockDim.x) {
    float sc = dcol[(size_t)c * 16] / (knorm * mn[c] + EPSV);
    s += __expf(sc * beta - mx);
  }
  s = blockRed(s, red, true);
  float invs = 1.0f / s;

  auto wi = [&](int c) -> float {    // w_interp(c), recomputed from L2-resident data
    c &= (NC - 1);
    float sc = dcol[(size_t)c * 16] / (knorm * mn[c] + EPSV);
    return gate * __expf(sc * beta - mx) * invs + (1.0f - gate) * pw[c];
  };

  float sharp[NC / 1024];
  float s2a = 0.0f;
#pragma unroll
  for (int i = 0; i < NC / 1024; ++i) {
    int c = threadIdx.x + i * 1024;
    float v = s0 * wi(c - 1) + s1 * wi(c) + s2 * wi(c + 1);
    float sp = __powf(v, gamma);
    sharp[i] = sp;
    s2a += sp;
  }
  s2a = blockRed(s2a, red, true);
  float inv2 = 1.0f / (s2a + EPSV);
#pragma unroll
  for (int i = 0; i < NC / 1024; ++i) {
    int c = threadIdx.x + i * 1024;
    dist[(size_t)(b * NH + h) * NC + c] = sharp[i] * inv2;
  }
}

// ---------------- kernel 5: read = dist x memory via WMMA from LDS -------------
__global__ void ntm_zero_out(float* __restrict__ out) {
  out[blockIdx.x * blockDim.x + threadIdx.x] = 0.0f;
}

// block = 256 threads (8 waves) handles (batch, 64-cell K chunk); memory rows
// staged once into LDS and consumed by all 8 w-tile waves.
__global__ void ntm_read_kernel(const float* __restrict__ mem,
                                const float* __restrict__ dist,
                                float* __restrict__ out) {
  __shared__ __align__(16) float smem[64 * LDSS];   // 33.8 KB
  int tid = threadIdx.x;
  int b  = blockIdx.x >> 8;            // 256 K-chunks per batch
  int cbeg = (blockIdx.x & 255) << 6;

  const float* gbase = mem + ((size_t)b * NC + cbeg) * WRD;
#pragma unroll
  for (int i = 0; i < 8; ++i) {
    int ci = tid + i * 256;
    int row = ci >> 5, ch = ci & 31;
    copy16_async(gbase + row * WRD + ch * 4, smem + row * LDSS + ch * 4);
  }
  async_join();

  int wt = tid >> 5, lane = tid & 31;      // wave = one 16-wide w-tile
  int w0 = wt * 16;
  int lm = lane & 15, hi = lane >> 4;

  const float* arow = dist + (size_t)(b * NH + (lm & 7)) * NC + cbeg + 2 * hi;
  v8f acc = {0.f, 0.f, 0.f, 0.f, 0.f, 0.f, 0.f, 0.f};
#pragma unroll 4
  for (int r = 0; r < 64; r += 4) {
    v2f a;
    if (lm < NH) a = *(const v2f*)(arow + r);
    else { a.x = 0.0f; a.y = 0.0f; }          // padded heads contribute zero
    int rr = r + 2 * hi;
    v2f bb;
    bb.x = smem[rr * LDSS + w0 + lm];
    bb.y = smem[(rr + 1) * LDSS + w0 + lm];
    acc = __builtin_amdgcn_wmma_f32_16x16x4_f32(false, a, false, bb,
                                                (short)0, acc, false, false);
  }
  // D rows M = v + 8*hi; only M<8 are real heads (hi==0 half-wave)
  if (hi == 0) {
#pragma unroll
    for (int v = 0; v < NH; ++v)
      atomicAdd(&out[(size_t)(b * NH + v) * WRD + w0 + lane], acc[v]);
  }
}

// ---------------- host ----------------
extern "C" void kernel_launch(void* const* d_in, const int* in_sizes, int n_in,
                              void* d_out, int out_size, void* d_ws, size_t ws_size,
                              hipStream_t stream) {
  const float* memory   = (const float*)d_in[0];   // [16,16384,128]
  const float* controls = (const float*)d_in[1];   // [16,1072]
  const float* bias     = (const float*)d_in[2];   // [1,8,16384]
  float* out = (float*)d_out;                      // [16,8,128]
  float* ws  = (float*)d_ws;

  float* keysT  = ws;                    // [16][16][128]      = 32768
  float* params = keysT + 32768;         // [16][8][8]         = 1024
  float* mnorm  = params + 1024;         // [16][16384]        = 262144
  float* dots   = mnorm + 262144;        // [16][16384][16]    = 4194304
  float* prevW  = dots + 4194304;        // [8][16384]         = 131072
  float* dist   = prevW + 131072;        // [16][8][16384]     = 2097152

  ntm_prep_kernel<<<1, 256, 0, stream>>>(controls, keysT, params);
  ntm_prevw_kernel<<<NH, 256, 0, stream>>>(bias, prevW);

  // 16 batches * 256 row-groups of 64 cells, 4 waves per block
  ntm_dots_kernel<<<NB * 256, 128, 0, stream>>>(memory, keysT, dots, mnorm);

  ntm_weights_kernel<<<NB * NH, 1024, 0, stream>>>(dots, mnorm, params, prevW, dist);

  ntm_zero_out<<<(NB * NH * WRD) / 256, 256, 0, stream>>>(out);

  // 16 batches * 256 K-chunks, 8 w-tile waves per block
  ntm_read_kernel<<<NB * 256, 256, 0, stream>>>(memory, dist, out);
}